// KnowledgeLayer_29686813950481
// MI455X (gfx1250) — compile-verified
//
#include <hip/hip_runtime.h>

// Fully-fused sum-product-network evaluation for MI455X (gfx1250).
//
//   h0 = encode(x)            [4098,1024]  (virtual, computed on the fly)
//   h1 = h0[idx0].prod(1)     [32768,1024] (virtual)
//   h2 = h1[idx1].sum(1)      [16384,1024] (virtual)
//   h3 = h2[idx2].prod(1)     [8192,1024]  (virtual)
//   out= h3[idx3].sum(1)      [4096,1024]
//
// Full fusion: each output row chases 15 int2 index pairs (all SGPR-uniform
// per block -> scalar s_load chase) down to 16 leaf rows of the encoded
// input, read directly from x (8 MB, L2-resident). ~272 MB total traffic vs
// ~720 MB layer-by-layer.
//
// Data movement (CDNA5 path): the 16 leaf-row segments are fetched with
// GLOBAL_LOAD_ASYNC_TO_LDS_B128 (ASYNCcnt), split-waited so the first
// subtree is consumed from LDS while the second half is still in flight.
// Falls back to direct global_load_b128 if the builtin is unavailable.

typedef float v4f __attribute__((ext_vector_type(4)));

#define BATCH 1024   // batch columns per row (fp32)
#define NLEAF 16

#if defined(__has_builtin)
#if __has_builtin(__builtin_amdgcn_global_load_async_to_lds_b128)
#define HAVE_ASYNC_LDS 1
#endif
#endif

#ifdef HAVE_ASYNC_LDS
// Builtin prototype (from compiler diagnostic): params are pointers to
// int __attribute__((vector_size(16))) in global (AS1) / LDS (AS3).
typedef int v4i __attribute__((vector_size(16)));
typedef __attribute__((address_space(1))) v4i* g_v4i_ptr;
typedef __attribute__((address_space(3))) v4i* l_v4i_ptr;
#endif

// ---- encode-on-the-fly helpers -------------------------------------------
// h0 row r:  r==0 -> 0, r==1 -> 1, even r>=2 -> x[(r-2)/2], odd -> 1-x[(r-2)>>1]
// r is SGPR-uniform within the block -> scalar branches.

__device__ __forceinline__ v4f enc_fix(v4f v, int r) {
    if (r >= 2) {
        if (r & 1) v = 1.0f - v;
        return v;
    }
    const float f = (float)r;               // 0 -> 0.0f, 1 -> 1.0f
    v4f k = {f, f, f, f};
    return k;
}

__device__ __forceinline__ v4f enc_load_direct(const float* __restrict__ x, int r, int c) {
    v4f v = {0.f, 0.f, 0.f, 0.f};
    if (r >= 2) {
        const size_t row = (size_t)((unsigned)(r - 2) >> 1);
        v = *reinterpret_cast<const v4f*>(x + row * BATCH + c);
    }
    return enc_fix(v, r);
}

extern "C" __global__ __launch_bounds__(256, 2)
void spn_fused_kernel(const float* __restrict__ x,
                      const int2*  __restrict__ idx0,
                      const int2*  __restrict__ idx1,
                      const int2*  __restrict__ idx2,
                      const int2*  __restrict__ idx3,
                      float*       __restrict__ out)
{
    const int j = (int)blockIdx.x;          // output row
    const int c = (int)(threadIdx.x << 2);  // 4 batch columns per lane

    // ---- Uniform index chase (15 scalar int2 loads, all SGPR) ----
    const int2 i  = idx3[j];
    const int2 c0 = idx2[i.x];
    const int2 c1 = idx2[i.y];
    const int2 s0 = idx1[c0.x];
    const int2 s1 = idx1[c0.y];
    const int2 s2 = idx1[c1.x];
    const int2 s3 = idx1[c1.y];
    const int2 p0 = idx0[s0.x];
    const int2 p1 = idx0[s0.y];
    const int2 p2 = idx0[s1.x];
    const int2 p3 = idx0[s1.y];
    const int2 p4 = idx0[s2.x];
    const int2 p5 = idx0[s2.y];
    const int2 p6 = idx0[s3.x];
    const int2 p7 = idx0[s3.y];

    const int leaf[NLEAF] = {p0.x, p0.y, p1.x, p1.y, p2.x, p2.y, p3.x, p3.y,
                             p4.x, p4.y, p5.x, p5.y, p6.x, p6.y, p7.x, p7.y};

#ifdef HAVE_ASYNC_LDS
    // ---- Async gather: 16 per-lane 16B copies, memory -> LDS (ASYNCcnt) ----
    __shared__ float stage[NLEAF * BATCH];  // 64 KB: 16 rows x 4 KB

#pragma unroll
    for (int k = 0; k < NLEAF; ++k) {
        const int r = leaf[k];
        // r<2 leaves have no backing row: copy a dummy (row 0) to keep the
        // ASYNCcnt schedule static; the value is overridden at consume time.
        const size_t row = (r >= 2) ? (size_t)((unsigned)(r - 2) >> 1) : 0;
        float* g = const_cast<float*>(x) + row * BATCH + c;
        float* l = stage + k * BATCH + c;
        __builtin_amdgcn_global_load_async_to_lds_b128(
            (g_v4i_ptr)g, (l_v4i_ptr)l, 0, 0);
    }

    // Async loads retire IN ORDER: cnt<=8 means leaves 0..7 have landed.
    asm volatile("s_wait_asynccnt 0x8" ::: "memory");
    v4f h1_0 = enc_fix(*reinterpret_cast<const v4f*>(stage + 0 * BATCH + c), leaf[0])
             * enc_fix(*reinterpret_cast<const v4f*>(stage + 1 * BATCH + c), leaf[1]);
    v4f h1_1 = enc_fix(*reinterpret_cast<const v4f*>(stage + 2 * BATCH + c), leaf[2])
             * enc_fix(*reinterpret_cast<const v4f*>(stage + 3 * BATCH + c), leaf[3]);
    v4f h1_2 = enc_fix(*reinterpret_cast<const v4f*>(stage + 4 * BATCH + c), leaf[4])
             * enc_fix(*reinterpret_cast<const v4f*>(stage + 5 * BATCH + c), leaf[5]);
    v4f h1_3 = enc_fix(*reinterpret_cast<const v4f*>(stage + 6 * BATCH + c), leaf[6])
             * enc_fix(*reinterpret_cast<const v4f*>(stage + 7 * BATCH + c), leaf[7]);
    const v4f h3_0 = (h1_0 + h1_1) * (h1_2 + h1_3);   // subtree of idx2[i.x]

    asm volatile("s_wait_asynccnt 0x0" ::: "memory");
    v4f h1_4 = enc_fix(*reinterpret_cast<const v4f*>(stage +  8 * BATCH + c), leaf[8])
             * enc_fix(*reinterpret_cast<const v4f*>(stage +  9 * BATCH + c), leaf[9]);
    v4f h1_5 = enc_fix(*reinterpret_cast<const v4f*>(stage + 10 * BATCH + c), leaf[10])
             * enc_fix(*reinterpret_cast<const v4f*>(stage + 11 * BATCH + c), leaf[11]);
    v4f h1_6 = enc_fix(*reinterpret_cast<const v4f*>(stage + 12 * BATCH + c), leaf[12])
             * enc_fix(*reinterpret_cast<const v4f*>(stage + 13 * BATCH + c), leaf[13]);
    v4f h1_7 = enc_fix(*reinterpret_cast<const v4f*>(stage + 14 * BATCH + c), leaf[14])
             * enc_fix(*reinterpret_cast<const v4f*>(stage + 15 * BATCH + c), leaf[15]);
    const v4f h3_1 = (h1_4 + h1_5) * (h1_6 + h1_7);   // subtree of idx2[i.y]

    const v4f r = h3_0 + h3_1;
#else
    // ---- Fallback: direct coalesced b128 gathers ----
    const v4f h1_0 = enc_load_direct(x, leaf[0],  c) * enc_load_direct(x, leaf[1],  c);
    const v4f h1_1 = enc_load_direct(x, leaf[2],  c) * enc_load_direct(x, leaf[3],  c);
    const v4f h1_2 = enc_load_direct(x, leaf[4],  c) * enc_load_direct(x, leaf[5],  c);
    const v4f h1_3 = enc_load_direct(x, leaf[6],  c) * enc_load_direct(x, leaf[7],  c);
    const v4f h1_4 = enc_load_direct(x, leaf[8],  c) * enc_load_direct(x, leaf[9],  c);
    const v4f h1_5 = enc_load_direct(x, leaf[10], c) * enc_load_direct(x, leaf[11], c);
    const v4f h1_6 = enc_load_direct(x, leaf[12], c) * enc_load_direct(x, leaf[13], c);
    const v4f h1_7 = enc_load_direct(x, leaf[14], c) * enc_load_direct(x, leaf[15], c);
    const v4f h3_0 = (h1_0 + h1_1) * (h1_2 + h1_3);
    const v4f h3_1 = (h1_4 + h1_5) * (h1_6 + h1_7);
    const v4f r = h3_0 + h3_1;
#endif

    // Output is write-once: non-temporal store keeps `x` resident in L2.
    __builtin_nontemporal_store(r, reinterpret_cast<v4f*>(out + (size_t)j * BATCH + c));
}

extern "C" void kernel_launch(void* const* d_in, const int* in_sizes, int n_in,
                              void* d_out, int out_size, void* d_ws, size_t ws_size,
                              hipStream_t stream) {
    const float* x    = (const float*)d_in[0];   // [2048, 1024] fp32
    const int2*  idx0 = (const int2*)d_in[1];    // [32768, 2]
    const int2*  idx1 = (const int2*)d_in[2];    // [16384, 2]
    const int2*  idx2 = (const int2*)d_in[3];    // [8192, 2]
    const int2*  idx3 = (const int2*)d_in[4];    // [4096, 2]
    float*       out  = (float*)d_out;           // [4096, 1024] fp32

    const int n_out_rows = in_sizes[4] / 2;      // 4096
    dim3 grid((unsigned)n_out_rows);
    dim3 block(BATCH / 4);                       // 256 threads = 8 wave32 waves

    hipLaunchKernelGGL(spn_fused_kernel, grid, block, 0, stream,
                       x, idx0, idx1, idx2, idx3, out);
}